// T2I_OptimalTransportAligner_27831388078754
// MI455X (gfx1250) — compile-verified
//
#include <hip/hip_runtime.h>
#include <hip/hip_bf16.h>
#include <math.h>

// ---------------------------------------------------------------------------
// T2I OptimalTransportAligner on MI455X (gfx1250), f32 WMMA path.
// Dims: B=8, N=1024, C=256 -> L=B*C=2048 ; M=96*96=9216
// GEMM tiling: block tile 128x64, 4 waves (2x2), wave tile 64x32,
// LDS tiles stored [row][k] with even pitch so each WMMA fragment is one
// ds_load_b64 (k and k+1 contiguous, 8B aligned).
// ---------------------------------------------------------------------------

typedef float v2f __attribute__((ext_vector_type(2)));
typedef float v8f __attribute__((ext_vector_type(8)));

static constexpr int kB = 8;
static constexpr int kN = 1024;
static constexpr int kC = 256;
static constexpr int kL = 2048;   // B*C
static constexpr int kM = 9216;   // 96*96
static constexpr int KP = 18;     // LDS pitch in floats ([row][k] layout), even -> b64 aligned

static __device__ __forceinline__ v8f wmma_f32(v2f a, v2f b, v8f c) {
  // V_WMMA_F32_16X16X4_F32 : D(16x16,f32) = A(16x4,f32) x B(4x16,f32) + C
  return __builtin_amdgcn_wmma_f32_16x16x4_f32(false, a, false, b, (short)0, c,
                                               false, false);
}

static __device__ __forceinline__ v2f lds_frag(const float* p) {
  return *reinterpret_cast<const v2f*>(p);
}

// ---------------- prep kernels ----------------

// A_ws[l*1024 + n] = text[b, n, c]  (l = b*256 + c)
__global__ __launch_bounds__(256) void transposeA_kernel(
    const float* __restrict__ text, float* __restrict__ A_ws) {
  const int n = blockIdx.x * 256 + threadIdx.x;
  const int l = blockIdx.y;
  const int b = l >> 8, c = l & 255;
  A_ws[(size_t)l * kN + n] = text[(size_t)b * (kN * kC) + (size_t)n * kC + c];
}

// sqA[n] = sum_l A^T[n,l]^2 ; also u[n] = 1
__global__ __launch_bounds__(256) void sqA_kernel(const float* __restrict__ text,
                                                  float* __restrict__ sqA,
                                                  float* __restrict__ u) {
  __shared__ float red[256];
  const int n = blockIdx.x;
  const int c = threadIdx.x;
  float s = 0.f;
#pragma unroll
  for (int b = 0; b < kB; ++b) {
    float x = text[(size_t)b * (kN * kC) + (size_t)n * kC + c];
    s += x * x;
  }
  red[c] = s;
  __syncthreads();
  for (int off = 128; off > 0; off >>= 1) {
    if (c < off) red[c] += red[c + off];
    __syncthreads();
  }
  if (c == 0) { sqA[n] = red[0]; u[n] = 1.0f; }
}

// sqB[m] = sum_l Bf[l,m]^2 ; also v[m] = 1
__global__ __launch_bounds__(256) void sqB_kernel(const float* __restrict__ img,
                                                  float* __restrict__ sqB,
                                                  float* __restrict__ v) {
  const int m = blockIdx.x * 256 + threadIdx.x;
  float s = 0.f;
  for (int l = 0; l < kL; ++l) {
    float x = img[(size_t)l * kM + m];
    s += x * x;
  }
  sqB[m] = s;
  v[m] = 1.0f;
}

__global__ void init_flags_kernel(int* done) { *done = 0; }

// ---------------- shared GEMM core -----------------------------------------
// lds_a: [128][KP] rows of the output-M dim; lds_b: [64][KP] cols (output-N).
// Wave w: wm = (w>>1)*64, wn = (w&1)*32. acc[i][j]: rows wm+16i.., cols wn+16j..
// Per k-step (K=4): 4 A frags + 2 B frags (ds_load_b64 each), 8 WMMAs.

#define GEMM_CORE_COMPUTE()                                                   \
  _Pragma("unroll") for (int ks = 0; ks < 4; ++ks) {                          \
    const int kb = ks * 4 + 2 * lhalf;                                        \
    v2f af[4], bf[2];                                                         \
    _Pragma("unroll") for (int i = 0; i < 4; ++i)                             \
        af[i] = lds_frag(&lds_a[(wm + 16 * i + lmod) * KP + kb]);             \
    _Pragma("unroll") for (int j = 0; j < 2; ++j)                             \
        bf[j] = lds_frag(&lds_b[(wn + 16 * j + lmod) * KP + kb]);             \
    _Pragma("unroll") for (int i = 0; i < 4; ++i)                             \
        _Pragma("unroll") for (int j = 0; j < 2; ++j)                         \
            acc[i][j] = wmma_f32(af[i], bf[j], acc[i][j]);                    \
  }

// ---------------- GEMM 1: cost/K  (A^T @ Bf with distance epilogue) ---------

__global__ __launch_bounds__(128) void gemm_cost_kernel(
    const float* __restrict__ A, const float* __restrict__ Bimg,
    const float* __restrict__ sqA, const float* __restrict__ sqB,
    float* __restrict__ cost, float* __restrict__ Kmat) {
  __shared__ float lds_a[128 * KP];
  __shared__ float lds_b[64 * KP];
  const int tid = threadIdx.x;
  const int lane = tid & 31, wave = tid >> 5;
  const int lhalf = lane >> 4, lmod = lane & 15;
  const int wm = (wave >> 1) * 64, wn = (wave & 1) * 32;
  const int r0 = blockIdx.y * 128;  // n (rows of cost)
  const int c0 = blockIdx.x * 64;   // m (cols of cost)

  v8f acc[4][2];
#pragma unroll
  for (int i = 0; i < 4; ++i)
#pragma unroll
    for (int j = 0; j < 2; ++j) acc[i][j] = (v8f)0.0f;

  for (int k0 = 0; k0 < kL; k0 += 16) {
    // A^T(n,l) = A_ws[l*N+n] : contiguous in n -> m-coalesced reads
    for (int t = tid; t < 128 * 16; t += 128) {
      const int m = t & 127, k = t >> 7;
      lds_a[m * KP + k] = A[(size_t)(k0 + k) * kN + r0 + m];
    }
    for (int t = tid; t < 64 * 16; t += 128) {
      const int c = t & 63, k = t >> 6;
      lds_b[c * KP + k] = Bimg[(size_t)(k0 + k) * kM + c0 + c];
    }
    __syncthreads();
    GEMM_CORE_COMPUTE();
    __syncthreads();
  }

#pragma unroll
  for (int i = 0; i < 4; ++i)
#pragma unroll
    for (int j = 0; j < 2; ++j)
#pragma unroll
      for (int r = 0; r < 8; ++r) {
        const int gr = r0 + wm + 16 * i + 8 * lhalf + r;
        const int gc = c0 + wn + 16 * j + lmod;
        const float g = acc[i][j][r];
        const float sq = sqA[gr] + sqB[gc] - 2.0f * g;
        const float d = sqrtf(fmaxf(sq, 1e-12f));
        const size_t o = (size_t)gr * kM + gc;
        cost[o] = d;
        Kmat[o] = expf(d * -10.0f);  // exp(-cost/EPSILON)
      }
}

// ---------------- Sinkhorn iterations ----------------

__global__ __launch_bounds__(256) void sink_row_kernel(
    const float* __restrict__ Kmat, const float* __restrict__ v,
    float* __restrict__ u, float* __restrict__ du, const int* __restrict__ done) {
  if (*done) return;
  __shared__ float red[256];
  const int i = blockIdx.x;
  const float* row = Kmat + (size_t)i * kM;
  float s = 0.f;
  for (int j = threadIdx.x; j < kM; j += 256) s += row[j] * v[j];
  red[threadIdx.x] = s;
  __syncthreads();
  for (int off = 128; off > 0; off >>= 1) {
    if (threadIdx.x < off) red[threadIdx.x] += red[threadIdx.x + off];
    __syncthreads();
  }
  if (threadIdx.x == 0) {
    const float un = 1.0f / (red[0] + 100.0f);
    const float d = un - u[i];
    du[i] = d * d;
    u[i] = un;
  }
}

__global__ __launch_bounds__(256) void sink_col_kernel(
    const float* __restrict__ Kmat, const float* __restrict__ u,
    float* __restrict__ v, float* __restrict__ dv, const int* __restrict__ done) {
  if (*done) return;
  __shared__ float us[kN];
  for (int i = threadIdx.x; i < kN; i += 256) us[i] = u[i];
  __syncthreads();
  const int j = blockIdx.x * 256 + threadIdx.x;
  float s = 0.f;
  for (int i = 0; i < kN; ++i) s += Kmat[(size_t)i * kM + j] * us[i];
  const float vn = 1.0f / (s + 100.0f);
  const float d = vn - v[j];
  dv[j] = d * d;
  v[j] = vn;
}

__global__ __launch_bounds__(256) void sink_check_kernel(
    const float* __restrict__ du, const float* __restrict__ dv, int* done) {
  if (*done) return;
  __shared__ float red[256];
  const int tid = threadIdx.x;
  float s = 0.f, t = 0.f;
  for (int i = tid; i < kN; i += 256) s += du[i];
  for (int j = tid; j < kM; j += 256) t += dv[j];
  red[tid] = s;
  __syncthreads();
  for (int off = 128; off > 0; off >>= 1) {
    if (tid < off) red[tid] += red[tid + off];
    __syncthreads();
  }
  const float nu = red[0];
  __syncthreads();
  red[tid] = t;
  __syncthreads();
  for (int off = 128; off > 0; off >>= 1) {
    if (tid < off) red[tid] += red[tid + off];
    __syncthreads();
  }
  if (tid == 0) {
    // CONV_TOL = 0.01 -> squared-norm tol = 1e-4
    if (nu < 1e-4f && red[0] < 1e-4f) *done = 1;
  }
}

// T[i,j] = exp(-cost*EPSILON) * u[i] * v[j]  (overwrites K buffer)
__global__ __launch_bounds__(256) void make_T_kernel(
    const float* __restrict__ cost, float* __restrict__ T,
    const float* __restrict__ u, const float* __restrict__ v) {
  const int j = blockIdx.x * 256 + threadIdx.x;
  const int i = blockIdx.y;
  const size_t o = (size_t)i * kM + j;
  T[o] = expf(cost[o] * -0.1f) * u[i] * v[j];
}

// ---------------- GEMM 2: aligned_B = A @ T  -> image output ----------------

__global__ __launch_bounds__(128) void gemm_alignedB_kernel(
    const float* __restrict__ A, const float* __restrict__ T,
    float* __restrict__ outImg) {
  __shared__ float lds_a[128 * KP];
  __shared__ float lds_b[64 * KP];
  const int tid = threadIdx.x;
  const int lane = tid & 31, wave = tid >> 5;
  const int lhalf = lane >> 4, lmod = lane & 15;
  const int wm = (wave >> 1) * 64, wn = (wave & 1) * 32;
  const int r0 = blockIdx.y * 128;  // l
  const int c0 = blockIdx.x * 64;   // m

  v8f acc[4][2];
#pragma unroll
  for (int i = 0; i < 4; ++i)
#pragma unroll
    for (int j = 0; j < 2; ++j) acc[i][j] = (v8f)0.0f;

  for (int k0 = 0; k0 < kN; k0 += 16) {
    // A(l,n): contiguous in n (=k) -> 16-wide coalesced reads, contiguous LDS writes
    for (int t = tid; t < 128 * 16; t += 128) {
      const int k = t & 15, m = t >> 4;
      lds_a[m * KP + k] = A[(size_t)(r0 + m) * kN + k0 + k];
    }
    for (int t = tid; t < 64 * 16; t += 128) {
      const int c = t & 63, k = t >> 6;
      lds_b[c * KP + k] = T[(size_t)(k0 + k) * kM + c0 + c];  // T(n,m)
    }
    __syncthreads();
    GEMM_CORE_COMPUTE();
    __syncthreads();
  }

#pragma unroll
  for (int i = 0; i < 4; ++i)
#pragma unroll
    for (int j = 0; j < 2; ++j)
#pragma unroll
      for (int r = 0; r < 8; ++r) {
        const int gr = r0 + wm + 16 * i + 8 * lhalf + r;  // l
        const int gc = c0 + wn + 16 * j + lmod;           // m
        outImg[(size_t)gr * kM + gc] = acc[i][j][r];
      }
}

// ------------- GEMM 3: aligned_A = Bf @ T^T -> text output (scatter) --------

__global__ __launch_bounds__(128) void gemm_alignedA_kernel(
    const float* __restrict__ img, const float* __restrict__ T,
    float* __restrict__ outText) {
  __shared__ float lds_a[128 * KP];
  __shared__ float lds_b[64 * KP];
  const int tid = threadIdx.x;
  const int lane = tid & 31, wave = tid >> 5;
  const int lhalf = lane >> 4, lmod = lane & 15;
  const int wm = (wave >> 1) * 64, wn = (wave & 1) * 32;
  const int r0 = blockIdx.y * 128;  // l
  const int c0 = blockIdx.x * 64;   // n

  v8f acc[4][2];
#pragma unroll
  for (int i = 0; i < 4; ++i)
#pragma unroll
    for (int j = 0; j < 2; ++j) acc[i][j] = (v8f)0.0f;

  for (int k0 = 0; k0 < kM; k0 += 16) {
    for (int t = tid; t < 128 * 16; t += 128) {
      const int k = t & 15, m = t >> 4;
      lds_a[m * KP + k] = img[(size_t)(r0 + m) * kM + k0 + k];  // Bf(l,m)
    }
    for (int t = tid; t < 64 * 16; t += 128) {
      const int k = t & 15, c = t >> 4;
      lds_b[c * KP + k] = T[(size_t)(c0 + c) * kM + k0 + k];  // T^T(m,n)=T[n*M+m]
    }
    __syncthreads();
    GEMM_CORE_COMPUTE();
    __syncthreads();
  }

#pragma unroll
  for (int i = 0; i < 4; ++i)
#pragma unroll
    for (int j = 0; j < 2; ++j)
#pragma unroll
      for (int r = 0; r < 8; ++r) {
        const int gr = r0 + wm + 16 * i + 8 * lhalf + r;  // l = b*256 + c
        const int gc = c0 + wn + 16 * j + lmod;           // n
        const int b = gr >> 8, ch = gr & 255;
        outText[(size_t)b * (kN * kC) + (size_t)gc * kC + ch] = acc[i][j][r];
      }
}

// ---------------------------------------------------------------------------

extern "C" void kernel_launch(void* const* d_in, const int* in_sizes, int n_in,
                              void* d_out, int out_size, void* d_ws, size_t ws_size,
                              hipStream_t stream) {
  const float* text = (const float*)d_in[0];  // [8,1024,256]
  const float* img  = (const float*)d_in[1];  // [8,256,96,96] == Bf[2048,9216]
  float* out = (float*)d_out;                 // text(2097152) ++ image(18874368)

  // workspace layout (floats)
  float* A_ws    = (float*)d_ws;            // 2048*1024
  float* cost_ws = A_ws + (size_t)kL * kN;  // 1024*9216
  float* K_ws    = cost_ws + (size_t)kN * kM;  // 1024*9216 (reused as T)
  float* sqA = K_ws + (size_t)kN * kM;      // 1024
  float* sqB = sqA + kN;                    // 9216
  float* u   = sqB + kM;                    // 1024
  float* v   = u + kN;                      // 9216
  float* du  = v + kM;                      // 1024
  float* dv  = du + kN;                     // 9216
  int* done  = (int*)(dv + kM);

  float* outText = out;
  float* outImg  = out + (size_t)kB * kN * kC;

  // prep
  transposeA_kernel<<<dim3(kN / 256, kL), 256, 0, stream>>>(text, A_ws);
  sqA_kernel<<<dim3(kN), 256, 0, stream>>>(text, sqA, u);
  sqB_kernel<<<dim3(kM / 256), 256, 0, stream>>>(img, sqB, v);
  init_flags_kernel<<<1, 1, 0, stream>>>(done);

  // cost + K
  gemm_cost_kernel<<<dim3(kM / 64, kN / 128), 128, 0, stream>>>(A_ws, img, sqA,
                                                                sqB, cost_ws, K_ws);

  // Sinkhorn, 50 iterations with convergence freeze
  for (int it = 0; it < 50; ++it) {
    sink_row_kernel<<<dim3(kN), 256, 0, stream>>>(K_ws, v, u, du, done);
    sink_col_kernel<<<dim3(kM / 256), 256, 0, stream>>>(K_ws, u, v, dv, done);
    sink_check_kernel<<<1, 256, 0, stream>>>(du, dv, done);
  }

  // transport plan (overwrites K buffer)
  make_T_kernel<<<dim3(kM / 256, kN), 256, 0, stream>>>(cost_ws, K_ws, u, v);

  // aligned outputs
  gemm_alignedB_kernel<<<dim3(kM / 64, kL / 128), 128, 0, stream>>>(A_ws, K_ws, outImg);
  gemm_alignedA_kernel<<<dim3(kN / 64, kL / 128), 128, 0, stream>>>(img, K_ws, outText);
}